// SelfAttention_6133213299212
// MI455X (gfx1250) — compile-verified
//
#include <hip/hip_runtime.h>
#include <math.h>

#define H 8
#define DH 64
#define B 8
#define L 512
#define LL (L * L)
#define BN_EPS 1e-5f

#define MT 128        // Q rows per block in attention
#define JT 64         // K/V rows per tile
#define KSTR 68       // padded LDS row stride (floats): row start = 272B, 16B aligned
#define QSTR KSTR
#define SSTR 18

typedef float v2f __attribute__((ext_vector_type(2)));
typedef float v8f __attribute__((ext_vector_type(8)));
typedef int   v4i __attribute__((ext_vector_type(4)));
typedef __attribute__((address_space(1))) v4i* gv4i_p;   // global v4i*
typedef __attribute__((address_space(3))) v4i* lv4i_p;   // LDS v4i*

#if __has_builtin(__builtin_amdgcn_global_load_async_to_lds_b128) && \
    __has_builtin(__builtin_amdgcn_s_wait_asynccnt)
#define HAVE_ASYNC 1
#define WAIT_ASYNC(n) __builtin_amdgcn_s_wait_asynccnt(n)
#else
#define HAVE_ASYNC 0
#define WAIT_ASYNC(n)
#endif

// ---------------- ws layout (floats) ----------------
// [0..7]   ch_sum      [8..15]  ch_sumsq    [16..79] bh_sum
// [80..87] mean        [88..95] invstd      [96..159] se_scale
// [160..160+1023*8) pe_table

__global__ void zero_stats_kernel(float* ws) {
    int t = threadIdx.x;
    if (t < 80) ws[t] = 0.0f;
}

// Conv 3x3 SAME (NCHW, OIHW) + bias -> x (stored in bpp region), accumulate stats.
__global__ void conv_stats_kernel(const float* __restrict__ structure,
                                  const float* __restrict__ conv_w,
                                  const float* __restrict__ conv_b,
                                  float* __restrict__ xout,
                                  float* __restrict__ ch_sum,
                                  float* __restrict__ ch_sumsq,
                                  float* __restrict__ bh_sum) {
    __shared__ float wsh[H * 9];
    __shared__ float red[256];
    __shared__ float red2[256];
    const int tid = threadIdx.x;
    const int bh = blockIdx.x >> 10;      // 1024 pixel-chunks per (b,h) image
    const int chunk = blockIdx.x & 1023;
    const int b = bh >> 3, o = bh & 7;
    if (tid < H * 9) wsh[tid] = conv_w[o * H * 9 + tid];
    __syncthreads();
    const int pix = chunk * 256 + tid;
    const int i = pix >> 9, j = pix & 511;
    const float* img = structure + (size_t)b * H * LL;
    float acc = conv_b[o];
    for (int c = 0; c < H; ++c) {
        const float* p = img + (size_t)c * LL;
        #pragma unroll
        for (int di = -1; di <= 1; ++di) {
            int ii = i + di;
            if (ii < 0 || ii >= L) continue;
            #pragma unroll
            for (int dj = -1; dj <= 1; ++dj) {
                int jj = j + dj;
                if (jj < 0 || jj >= L) continue;
                acc += p[ii * L + jj] * wsh[c * 9 + (di + 1) * 3 + (dj + 1)];
            }
        }
    }
    xout[(size_t)bh * LL + pix] = acc;
    red[tid] = acc;
    red2[tid] = acc * acc;
    __syncthreads();
    for (int s = 128; s > 0; s >>= 1) {
        if (tid < s) { red[tid] += red[tid + s]; red2[tid] += red2[tid + s]; }
        __syncthreads();
    }
    if (tid == 0) {
        atomicAdd(&ch_sum[o], red[0]);
        atomicAdd(&ch_sumsq[o], red2[0]);
        atomicAdd(&bh_sum[bh], red[0]);
    }
}

// PE MLP over the 1023 distinct (j-i) values.  pe_table[r][h], r = (j-i)+511.
__global__ void pe_kernel(const float* __restrict__ pe_w1, const float* __restrict__ pe_b1,
                          const float* __restrict__ pe_w2, const float* __restrict__ pe_b2,
                          const float* __restrict__ pe_w3, const float* __restrict__ pe_b3,
                          float* __restrict__ pe_table) {
    int r = blockIdx.x * 128 + threadIdx.x;
    if (r >= 2 * L - 1) return;
    float rel = (float)(r - (L - 1));
    float h1[48], h2[48];
    #pragma unroll
    for (int k = 0; k < 48; ++k) {
        float a = rel * pe_w1[k] + pe_b1[k];
        h1[k] = a / (1.0f + expf(-a));        // silu
    }
    for (int k = 0; k < 48; ++k) {
        float a = pe_b2[k];
        #pragma unroll
        for (int m = 0; m < 48; ++m) a += pe_w2[k * 48 + m] * h1[m];
        h2[k] = a / (1.0f + expf(-a));
    }
    for (int o = 0; o < H; ++o) {
        float a = pe_b3[o];
        #pragma unroll
        for (int m = 0; m < 48; ++m) a += pe_w3[o * 48 + m] * h2[m];
        pe_table[r * H + o] = a;
    }
}

// Finalize BN stats + SE MLP.  One block of 64 threads (t = b*8+h).
__global__ void stats_se_kernel(const float* __restrict__ bn_gamma,
                                const float* __restrict__ bn_beta,
                                const float* __restrict__ se_w1, const float* __restrict__ se_b1,
                                const float* __restrict__ se_w2, const float* __restrict__ se_b2,
                                float* __restrict__ ws) {
    __shared__ float mean_s[H], inv_s[H];
    __shared__ float pooled[B][H], hid[B][H];
    const int t = threadIdx.x;
    const float* ch_sum = ws;
    const float* ch_sumsq = ws + 8;
    const float* bh_sum = ws + 16;
    float* mean = ws + 80;
    float* invstd = ws + 88;
    float* se_scale = ws + 96;
    if (t < H) {
        float n = (float)B * (float)LL;
        float m = ch_sum[t] / n;
        float v = ch_sumsq[t] / n - m * m;
        mean_s[t] = m;
        inv_s[t] = rsqrtf(v + BN_EPS);
        mean[t] = m;
        invstd[t] = inv_s[t];
    }
    __syncthreads();
    const int b = t >> 3, h = t & 7;
    // BN is affine => spatial mean of BN(x) = BN(spatial mean of x)
    pooled[b][h] = (bh_sum[t] / (float)LL - mean_s[h]) * inv_s[h] * bn_gamma[h] + bn_beta[h];
    __syncthreads();
    float a = se_b1[h];
    for (int k = 0; k < H; ++k) a += se_w1[h * H + k] * pooled[b][k];
    hid[b][h] = fmaxf(a, 0.0f);
    __syncthreads();
    float c = se_b2[h];
    for (int k = 0; k < H; ++k) c += se_w2[h * H + k] * hid[b][k];
    se_scale[t] = 1.0f / (1.0f + expf(-c));
}

// bpp = gelu_exact(BN(x)*se_scale) + structure, in-place over the bpp region.
__global__ void bn_se_gelu_kernel(const float* __restrict__ structure,
                                  const float* __restrict__ bn_gamma,
                                  const float* __restrict__ bn_beta,
                                  const float* __restrict__ ws,
                                  float* __restrict__ bpp) {
    int idx = blockIdx.x * 256 + threadIdx.x;   // 16,777,216 elems < 2^31
    int bh = idx >> 18;                          // / LL
    int h = bh & 7;
    float x = bpp[idx];
    float y = (x - ws[80 + h]) * ws[88 + h] * bn_gamma[h] + bn_beta[h];
    y *= ws[96 + bh];
    float g = 0.5f * y * (1.0f + erff(y * 0.70710678118654752f));
    bpp[idx] = g + structure[idx];
}

// Stage one JT x 64 tile (rows j0..j0+63 of [L][512] view) into padded LDS.
// Each wave moves 8 rows (2 KB) = 4 b128 transfers; ASYNCcnt += 4 per wave.
__device__ __forceinline__ void stage_tile(float* dstbase, const float* qbase,
                                           int j0, int wave, int lane) {
    #pragma unroll
    for (int q = 0; q < 4; ++q) {
        int flat = q * 32 + lane;            // 0..127 within this wave's 8 rows
        int row = wave * 8 + (flat >> 4);    // 16 float4 per row
        int c4 = flat & 15;
        const float* src = qbase + (size_t)(j0 + row) * (H * DH) + c4 * 4;
        float* dst = dstbase + row * KSTR + c4 * 4;
#if HAVE_ASYNC
        __builtin_amdgcn_global_load_async_to_lds_b128((gv4i_p)src, (lv4i_p)dst, 0, 0);
#else
        *(float4*)dst = *(const float4*)src;
#endif
    }
}

// Fused attention: O = (Q K^T / 8 + pe + bpp) V  via V_WMMA_F32_16X16X4_F32,
// Q fragments register-resident, K/V tiles double-buffered with async-to-LDS.
__global__ __launch_bounds__(256) void attn_kernel(const float* __restrict__ seq,
                                                   const float* __restrict__ bpp,
                                                   const float* __restrict__ pe_table,
                                                   float* __restrict__ out) {
    // Q staging (128 x KSTR) and the two K/V ping-pong buffers (2 x 64 x KSTR)
    // occupy the same 34,816 B of LDS: Q is dead once fragments are in registers.
    __shared__ __attribute__((aligned(16))) float smemA[2 * JT * KSTR];
    __shared__ float Ss[8][16][SSTR];

    const int tid = threadIdx.x;
    const int wave = tid >> 5;
    const int lane = tid & 31;
    const int ln = lane & 15;       // n / m within half
    const int half = lane >> 4;     // K-half selector in A/B fragments

    const int blk = blockIdx.x;
    const int bh = blk >> 2;        // 4 row-blocks per (b,h)
    const int i0 = (blk & 3) * MT;
    const int b = bh >> 3, h = bh & 7;

    const float* qbase = seq + (size_t)b * L * (H * DH) + h * DH;  // [l][512] view
    const float* bpp_bh = bpp + (size_t)bh * LL;

    // ---- phase 1: stage Q tile (MT x 64), preload A-fragments to registers ----
    {
        int r = tid >> 1;
        int cq = (tid & 1) * 8;
        const float4* src = (const float4*)(qbase + (size_t)(i0 + r) * (H * DH)) + cq;
        float4* dst = (float4*)(smemA + r * QSTR) + cq;
        #pragma unroll
        for (int q = 0; q < 8; ++q) dst[q] = src[q];
    }
    __syncthreads();

    const int wrow = wave * 16;
    v2f qa[16];                     // A 16x4 frags: lane=16*half+m holds Q[m][4kk+2half+{0,1}]
    #pragma unroll
    for (int kk = 0; kk < 16; ++kk)
        qa[kk] = *(const v2f*)(smemA + (wrow + ln) * QSTR + kk * 4 + half * 2);
    __syncthreads();                // Q buffer now dead -> becomes K/V ping-pong

    v8f acc[4];
    #pragma unroll
    for (int d = 0; d < 4; ++d)
        #pragma unroll
        for (int r = 0; r < 8; ++r) acc[d][r] = 0.0f;

    // prologue: kick off tile 0
    stage_tile(smemA, qbase, 0, wave, lane);

    for (int jt = 0; jt < L / JT; ++jt) {
        const int j0 = jt * JT;
        float* cur = smemA + (jt & 1) * (JT * KSTR);
        float* nxt = smemA + ((jt + 1) & 1) * (JT * KSTR);
        if (jt + 1 < L / JT) {
            stage_tile(nxt, qbase, j0 + JT, wave, lane);   // overlap with compute
            WAIT_ASYNC(4);          // tile jt complete (in-order), jt+1 outstanding
        } else {
            WAIT_ASYNC(0);
        }
        __syncthreads();            // publish all waves' portions of tile jt

        #pragma unroll
        for (int js = 0; js < 4; ++js) {
            // ---- S(16x16) = Q_wave(16x64) * K_sub^T(64x16) ----
            v8f c;
            #pragma unroll
            for (int r = 0; r < 8; ++r) c[r] = 0.0f;
            #pragma unroll
            for (int kk = 0; kk < 16; ++kk) {
                // B 4x16 = K^T: lane=16*half+n holds K[n][4kk+2half+{0,1}]
                v2f bb = *(const v2f*)(cur + (js * 16 + ln) * KSTR + kk * 4 + half * 2);
                c = __builtin_amdgcn_wmma_f32_16x16x4_f32(false, qa[kk], false, bb,
                                                          (short)0, c, false, false);
            }
            // ---- epilogue in C layout: VGPR r -> (M = r+8*half, N = ln) ----
            #pragma unroll
            for (int r = 0; r < 8; ++r) {
                int il = wrow + r + 8 * half;
                int gi = i0 + il;
                int gj = j0 + js * 16 + ln;
                float v = c[r] * 0.125f
                        + pe_table[(gj - gi + (L - 1)) * H + h]
                        + bpp_bh[(size_t)gi * L + gj];
                Ss[wave][r + 8 * half][ln] = v;     // per-wave scratch, same-wave reuse
            }
            // ---- O(16x64) += S(16x16) * V_sub(16x64) ----
            #pragma unroll
            for (int d = 0; d < 4; ++d) {
                #pragma unroll
                for (int jc = 0; jc < 4; ++jc) {
                    v2f a = *(const v2f*)&Ss[wave][ln][jc * 4 + half * 2];
                    v2f bb;
                    bb.x = cur[(js * 16 + jc * 4 + half * 2 + 0) * KSTR + d * 16 + ln];
                    bb.y = cur[(js * 16 + jc * 4 + half * 2 + 1) * KSTR + d * 16 + ln];
                    acc[d] = __builtin_amdgcn_wmma_f32_16x16x4_f32(false, a, false, bb,
                                                                   (short)0, acc[d], false, false);
                }
            }
        }
        __syncthreads();            // all waves done with `cur` before it is re-staged
    }

    // ---- store O: reference does a direct reshape of [B,H,L,64] ----
    float* obase = out + (size_t)bh * (L * DH) + (size_t)i0 * DH;
    #pragma unroll
    for (int d = 0; d < 4; ++d)
        #pragma unroll
        for (int r = 0; r < 8; ++r) {
            int il = wrow + r + 8 * half;
            obase[(size_t)il * DH + d * 16 + ln] = acc[d][r];
        }
}

extern "C" void kernel_launch(void* const* d_in, const int* in_sizes, int n_in,
                              void* d_out, int out_size, void* d_ws, size_t ws_size,
                              hipStream_t stream) {
    const float* seq    = (const float*)d_in[0];
    const float* stru   = (const float*)d_in[1];
    const float* conv_w = (const float*)d_in[2];
    const float* conv_b = (const float*)d_in[3];
    const float* bn_g   = (const float*)d_in[4];
    const float* bn_b   = (const float*)d_in[5];
    const float* se_w1  = (const float*)d_in[6];
    const float* se_b1  = (const float*)d_in[7];
    const float* se_w2  = (const float*)d_in[8];
    const float* se_b2  = (const float*)d_in[9];
    const float* pe_w1  = (const float*)d_in[10];
    const float* pe_b1  = (const float*)d_in[11];
    const float* pe_w2  = (const float*)d_in[12];
    const float* pe_b2  = (const float*)d_in[13];
    const float* pe_w3  = (const float*)d_in[14];
    const float* pe_b3  = (const float*)d_in[15];

    float* out_p = (float*)d_out;                       // [B,L,512] = 2,097,152 floats
    float* bpp_p = out_p + (size_t)B * L * (H * DH);    // [B,H,L,L] = 16,777,216 floats
    float* wsf = (float*)d_ws;
    float* pe_table = wsf + 160;

    zero_stats_kernel<<<1, 128, 0, stream>>>(wsf);
    conv_stats_kernel<<<B * H * 1024, 256, 0, stream>>>(stru, conv_w, conv_b, bpp_p,
                                                        wsf, wsf + 8, wsf + 16);
    pe_kernel<<<8, 128, 0, stream>>>(pe_w1, pe_b1, pe_w2, pe_b2, pe_w3, pe_b3, pe_table);
    stats_se_kernel<<<1, 64, 0, stream>>>(bn_g, bn_b, se_w1, se_b1, se_w2, se_b2, wsf);
    bn_se_gelu_kernel<<<(B * H * LL) / 256, 256, 0, stream>>>(stru, bn_g, bn_b, wsf, bpp_p);
    attn_kernel<<<B * H * (L / MT), 256, 0, stream>>>(seq, bpp_p, pe_table, out_p);
}